// Layer_87960930222765
// MI455X (gfx1250) — compile-verified
//
#include <hip/hip_runtime.h>

// ---------------------------------------------------------------------------
// FDFD Ez solver: dense complex LU (blocked, WMMA f32 trailing updates)
//   A (6656x6656 complex64, split f32 planes) x = b, 8 RHS
// GEMMs: LDS-staged via gfx1250 async copy (GLOBAL_LOAD_ASYNC_TO_LDS_B128),
//        4x v_wmma_f32_16x16x4_f32 per k-step (complex arithmetic).
// ---------------------------------------------------------------------------

#define NXg   128
#define NYg   52
#define NTOT  (NXg * NYg)          // 6656 = 64 * 104
#define NPMLg 10
#define NB    64                   // LU panel width
#define NBLK  (NTOT / NB)          // 104
#define NRHS  8

#define USE_ASYNC_STAGING 1

static constexpr double kPI   = 3.14159265358979323846;
static constexpr double kDL   = 2.5e-8;
static constexpr double kFREQ = 2.0e14;
static constexpr double kEPS0 = 8.85418782e-12;
static constexpr double kMU0  = 4.0e-7 * kPI;
static constexpr double kOMEGA = 2.0 * kPI * kFREQ;

typedef __attribute__((ext_vector_type(2))) float v2f;
typedef __attribute__((ext_vector_type(8))) float v8f;

__device__ inline v8f wmma4(v2f a, v2f b, v8f c) {
    return __builtin_amdgcn_wmma_f32_16x16x4_f32(false, a, false, b,
                                                 (short)0, c, false, false);
}

// Async global->LDS copy of 16 bytes (CDNA5 GLOBAL_LOAD_ASYNC_TO_LDS_B128,
// tracked by ASYNCcnt).  lds_dst must be a __shared__ address (low 32 bits of
// the generic address are the LDS offset); both addresses 16B-aligned.
__device__ inline void async_ldg128(void* lds_dst, const void* gsrc) {
#if USE_ASYNC_STAGING
    unsigned lo = (unsigned)(unsigned long long)lds_dst;
    asm volatile("global_load_async_to_lds_b128 %0, %1, off"
                 :: "v"(lo), "v"(gsrc) : "memory");
#else
    *(float4*)lds_dst = *(const float4*)gsrc;
#endif
}

__device__ inline void async_wait_all() {
#if USE_ASYNC_STAGING
    asm volatile("s_wait_asynccnt 0" ::: "memory");
#endif
}

// ---------------------------------------------------------------------------
// SC-PML stretch factor s = 1 - i*sigma/(omega*eps0)
// ---------------------------------------------------------------------------
__device__ inline void sfac(int n, int i, bool forward, double& sr, double& si) {
    const double m    = 3.0;
    const double dpml = NPMLg * kDL;
    const double eta0 = sqrt(kMU0 / kEPS0);
    const double smax = -(m + 1.0) * (-18.420680743952367) / (2.0 * eta0 * dpml); // -log(1e-8)
    double pos = (double)i + (forward ? 0.5 : 0.0);
    double dlo = fmax((double)NPMLg - pos, 0.0);
    double dhi = fmax(pos - (double)(n - 1 - NPMLg), 0.0);
    double d   = fmax(dlo, dhi) * kDL;
    double t   = d / dpml;
    double sigma = smax * t * t * t;
    sr = 1.0;
    si = -sigma / (kOMEGA * kEPS0);
}

__device__ inline void crecip2(double ar, double ai, double br, double bi,
                               double& rr, double& ri) {
    // 1 / (a*b)
    double pr = ar * br - ai * bi;
    double pi = ar * bi + ai * br;
    double den = pr * pr + pi * pi;
    rr =  pr / den;
    ri = -pi / den;
}

// ---------------------------------------------------------------------------
__global__ void zero_f(float* __restrict__ p, long n) {
    long i = (long)blockIdx.x * blockDim.x + threadIdx.x;
    if (i < n) p[i] = 0.0f;
}

// One thread per matrix row: scatter 5-point periodic stencil.
__global__ void build_A(float* __restrict__ Ar, float* __restrict__ Ai,
                        const float* __restrict__ rho,
                        const float* __restrict__ bg,
                        const float* __restrict__ opt) {
    int idx = blockIdx.x * blockDim.x + threadIdx.x;
    if (idx >= NTOT) return;
    int ix = idx / NYg, iy = idx % NYg;

    double sxf_r, sxf_i, sxb0_r, sxb0_i, sxb1_r, sxb1_i;
    double syf_r, syf_i, syb0_r, syb0_i, syb1_r, syb1_i;
    sfac(NXg, ix, true, sxf_r, sxf_i);
    sfac(NXg, (ix + 1) % NXg, false, sxb1_r, sxb1_i);
    sfac(NXg, ix, false, sxb0_r, sxb0_i);
    sfac(NYg, iy, true, syf_r, syf_i);
    sfac(NYg, (iy + 1) % NYg, false, syb1_r, syb1_i);
    sfac(NYg, iy, false, syb0_r, syb0_i);

    double cxp_r, cxp_i, cxm_r, cxm_i, cyp_r, cyp_i, cym_r, cym_i;
    crecip2(sxf_r, sxf_i, sxb1_r, sxb1_i, cxp_r, cxp_i);
    crecip2(sxf_r, sxf_i, sxb0_r, sxb0_i, cxm_r, cxm_i);
    crecip2(syf_r, syf_i, syb1_r, syb1_i, cyp_r, cyp_i);
    crecip2(syf_r, syf_i, syb0_r, syb0_i, cym_r, cym_i);

    float o  = opt[idx];
    float rt = o * rho[idx] + (1.0f - o) * bg[idx];
    rt = fminf(fmaxf(rt, 0.0f), 1.0f);
    double epsr = 1.0 + 11.0 * (double)rt;
    double k2 = (kOMEGA * kDL) * (kOMEGA * kDL) * kMU0 * kEPS0;

    double dg_r = -(cxp_r + cxm_r + cyp_r + cym_r) + k2 * epsr;
    double dg_i = -(cxp_i + cxm_i + cyp_i + cym_i);

    size_t row = (size_t)idx * NTOT;
    int cxp_c = ((ix + 1) % NXg) * NYg + iy;
    int cxm_c = ((ix + NXg - 1) % NXg) * NYg + iy;
    int cyp_c = ix * NYg + (iy + 1) % NYg;
    int cym_c = ix * NYg + (iy + NYg - 1) % NYg;

    Ar[row + cxp_c] += (float)cxp_r;  Ai[row + cxp_c] += (float)cxp_i;
    Ar[row + cxm_c] += (float)cxm_r;  Ai[row + cxm_c] += (float)cxm_i;
    Ar[row + cyp_c] += (float)cyp_r;  Ai[row + cyp_c] += (float)cyp_i;
    Ar[row + cym_c] += (float)cym_r;  Ai[row + cym_c] += (float)cym_i;
    Ar[row + idx]   += (float)dg_r;   Ai[row + idx]   += (float)dg_i;
}

// b = 1j*omega*mu0*dl^2 * (masks @ ics)  -> X holds RHS (solved in place)
__global__ void init_X(float* __restrict__ Xr, float* __restrict__ Xi,
                       const float* __restrict__ masks,
                       const float* __restrict__ ics) {
    int i = blockIdx.x * blockDim.x + threadIdx.x;
    if (i >= NRHS * NTOT) return;
    int b = i / NTOT, idx = i % NTOT;
    float s = masks[b * 2 + 0] * ics[idx] + masks[b * 2 + 1] * ics[NTOT + idx];
    const float c = (float)(kOMEGA * kMU0 * kDL * kDL);
    Xr[i] = 0.0f;
    Xi[i] = c * s;
}

// ---------------------------------------------------------------------------
// Complex GEMM: C = beta*C + alpha * A(MxK) * B(KxN), split re/im planes.
// Workgroup: 128 threads = 4 waves on a 32x32 output macro-tile.
// Per 64-wide K chunk: async-stage A(32x64) and B(64x32) into padded LDS,
// then each wave runs 16 fully-unrolled k-steps of 4 WMMA ops.
// All M,N multiples of 32, K multiple of 64; EXEC all-ones at every WMMA.
// ---------------------------------------------------------------------------
#define KT    64
#define LDA_S 68    // 32-row A tile, pad keeps 16B alignment + bank spread
#define LDB_S 36    // 64-row B tile

__global__ __launch_bounds__(128)
void cgemm_wmma(const float* __restrict__ Ar, const float* __restrict__ Ai, int lda,
                const float* __restrict__ Br, const float* __restrict__ Bi, int ldb,
                float* __restrict__ Cr, float* __restrict__ Ci, int ldc,
                int K, float alpha, int beta) {
    __shared__ __align__(16) float Asr[32 * LDA_S], Asi[32 * LDA_S];
    __shared__ __align__(16) float Bsr[KT * LDB_S], Bsi[KT * LDB_S];

    int t    = threadIdx.x;
    int wave = t >> 5;
    int lane = t & 31;
    int half = lane >> 4;      // 0: lanes 0-15, 1: lanes 16-31
    int m    = lane & 15;

    int rowBase = blockIdx.y * 32;
    int colBase = blockIdx.x * 32;
    int row0l   = (wave >> 1) * 16;   // wave's tile inside the 32x32 macro-tile
    int col0l   = (wave & 1) * 16;

    // Prefetch the C tile while we compute (lowers to global_prefetch_b8).
    if (beta && t < 32) {
        __builtin_prefetch(Cr + (size_t)(rowBase + t) * ldc + colBase, 0, 3);
        __builtin_prefetch(Ci + (size_t)(rowBase + t) * ldc + colBase, 0, 3);
    }

    v8f acc_rr = {}, acc_ii = {}, acc_ri = {}, acc_ir = {};

    for (int kk0 = 0; kk0 < K; kk0 += KT) {
        // ---- async stage A tile: 32 x 64 per plane = 512 x b128 ----
        for (int c = t; c < 512; c += 128) {
            int r = c >> 4, q = (c & 15) << 2;
            async_ldg128(&Asr[r * LDA_S + q], Ar + (size_t)(rowBase + r) * lda + kk0 + q);
            async_ldg128(&Asi[r * LDA_S + q], Ai + (size_t)(rowBase + r) * lda + kk0 + q);
        }
        // ---- async stage B tile: 64 x 32 per plane = 512 x b128 ----
        for (int c = t; c < 512; c += 128) {
            int r = c >> 3, q = (c & 7) << 2;
            async_ldg128(&Bsr[r * LDB_S + q], Br + (size_t)(kk0 + r) * ldb + colBase + q);
            async_ldg128(&Bsi[r * LDB_S + q], Bi + (size_t)(kk0 + r) * ldb + colBase + q);
        }
        async_wait_all();
        __syncthreads();

#pragma unroll
        for (int kk = 0; kk < KT; kk += 4) {
            int ka = kk + 2 * half;   // even -> 8B-aligned LDS pair loads
            // A 16x4 fragment: VGPR0 = K {0|2}, VGPR1 = K {1|3}
            v2f a_r, a_i, b_r, b_i;
            a_r.x = Asr[(row0l + m) * LDA_S + ka];
            a_r.y = Asr[(row0l + m) * LDA_S + ka + 1];
            a_i.x = Asi[(row0l + m) * LDA_S + ka];
            a_i.y = Asi[(row0l + m) * LDA_S + ka + 1];
            // B 4x16 fragment: VGPR j = rows K=j (lanes 0-15) / K=j+2 (16-31)
            b_r.x = Bsr[ka * LDB_S + col0l + m];
            b_r.y = Bsr[(ka + 1) * LDB_S + col0l + m];
            b_i.x = Bsi[ka * LDB_S + col0l + m];
            b_i.y = Bsi[(ka + 1) * LDB_S + col0l + m];

            acc_rr = wmma4(a_r, b_r, acc_rr);
            acc_ii = wmma4(a_i, b_i, acc_ii);
            acc_ri = wmma4(a_r, b_i, acc_ri);
            acc_ir = wmma4(a_i, b_r, acc_ir);
        }
        __syncthreads();   // LDS reuse across K chunks
    }

    // C/D 16x16 f32: VGPR v = row (v | v+8), col = lane%16
#pragma unroll
    for (int v = 0; v < 8; ++v) {
        int r = rowBase + row0l + v + 8 * half;
        int c = colBase + col0l + m;
        size_t off = (size_t)r * ldc + c;
        float cr = alpha * (acc_rr[v] - acc_ii[v]);
        float ci = alpha * (acc_ri[v] + acc_ir[v]);
        if (beta) { cr += Cr[off]; ci += Ci[off]; }
        Cr[off] = cr;
        Ci[off] = ci;
    }
}

// ---------------------------------------------------------------------------
// Factor 64x64 diagonal block in LDS (no pivoting), write L\U back into A,
// and emit inv(unit_lower) and inv(upper) blocks (re[4096] then im[4096]).
// ---------------------------------------------------------------------------
__global__ __launch_bounds__(256)
void lu_diag(float* __restrict__ Ar, float* __restrict__ Ai, int k0,
             float* __restrict__ Linv, float* __restrict__ Uinv) {
    __shared__ float Dr[NB * NB], Di[NB * NB];
    __shared__ float Wr[NB * NB], Wi[NB * NB];
    int tid = threadIdx.x;

    for (int e = tid; e < NB * NB; e += 256) {
        int r = e / NB, c = e % NB;
        size_t off = (size_t)(k0 + r) * NTOT + k0 + c;
        Dr[e] = Ar[off];
        Di[e] = Ai[off];
    }
    __syncthreads();

    // In-place LU of the block
    for (int j = 0; j < NB; ++j) {
        float pr = Dr[j * NB + j], pi = Di[j * NB + j];
        float den = pr * pr + pi * pi;
        float ivr = pr / den, ivi = -pi / den;
        for (int i = j + 1 + tid; i < NB; i += 256) {
            float xr = Dr[i * NB + j], xi = Di[i * NB + j];
            Dr[i * NB + j] = xr * ivr - xi * ivi;
            Di[i * NB + j] = xr * ivi + xi * ivr;
        }
        __syncthreads();
        int rem = NB - 1 - j;
        for (int e = tid; e < rem * rem; e += 256) {
            int i = j + 1 + e / rem, c = j + 1 + e % rem;
            float lr = Dr[i * NB + j], li = Di[i * NB + j];
            float ur = Dr[j * NB + c], ui = Di[j * NB + c];
            Dr[i * NB + c] -= lr * ur - li * ui;
            Di[i * NB + c] -= lr * ui + li * ur;
        }
        __syncthreads();
    }

    // Store factored block back to A
    for (int e = tid; e < NB * NB; e += 256) {
        int r = e / NB, c = e % NB;
        size_t off = (size_t)(k0 + r) * NTOT + k0 + c;
        Ar[off] = Dr[e];
        Ai[off] = Di[e];
    }

    // --- inv(U): one thread per column, backward substitution ---
    for (int e = tid; e < NB * NB; e += 256) { Wr[e] = 0.0f; Wi[e] = 0.0f; }
    __syncthreads();
    if (tid < NB) {
        int c = tid;
        float ur = Dr[c * NB + c], ui = Di[c * NB + c];
        float den = ur * ur + ui * ui;
        Wr[c * NB + c] = ur / den;
        Wi[c * NB + c] = -ui / den;
        for (int i = c - 1; i >= 0; --i) {
            float sr = 0.0f, si = 0.0f;
            for (int j = i + 1; j <= c; ++j) {
                float mr = Dr[i * NB + j], mi = Di[i * NB + j];
                float xr = Wr[j * NB + c], xi = Wi[j * NB + c];
                sr += mr * xr - mi * xi;
                si += mr * xi + mi * xr;
            }
            float dr = Dr[i * NB + i], di = Di[i * NB + i];
            float dden = dr * dr + di * di;
            float tr = -sr, ti = -si;
            Wr[i * NB + c] = (tr * dr + ti * di) / dden;
            Wi[i * NB + c] = (ti * dr - tr * di) / dden;
        }
    }
    __syncthreads();
    for (int e = tid; e < NB * NB; e += 256) {
        Uinv[e] = Wr[e];
        Uinv[NB * NB + e] = Wi[e];
    }
    __syncthreads();

    // --- inv(unit L): one thread per column, forward substitution ---
    for (int e = tid; e < NB * NB; e += 256) { Wr[e] = 0.0f; Wi[e] = 0.0f; }
    __syncthreads();
    if (tid < NB) {
        int c = tid;
        Wr[c * NB + c] = 1.0f;
        for (int i = c + 1; i < NB; ++i) {
            float sr = 0.0f, si = 0.0f;
            for (int j = c; j < i; ++j) {
                float mr = Dr[i * NB + j], mi = Di[i * NB + j];
                float xr = Wr[j * NB + c], xi = Wi[j * NB + c];
                sr += mr * xr - mi * xi;
                si += mr * xi + mi * xr;
            }
            Wr[i * NB + c] = -sr;
            Wi[i * NB + c] = -si;
        }
    }
    __syncthreads();
    for (int e = tid; e < NB * NB; e += 256) {
        Linv[e] = Wr[e];
        Linv[NB * NB + e] = Wi[e];
    }
}

// Copy temp panel (ld = lds_) into A region (ld = ldd)
__global__ void copy_panel(const float* __restrict__ sr, const float* __restrict__ si,
                           int lds_, float* __restrict__ dr, float* __restrict__ di,
                           int ldd, int rows, int cols) {
    int i = blockIdx.x * blockDim.x + threadIdx.x;
    if (i >= rows * cols) return;
    int r = i / cols, c = i % cols;
    dr[(size_t)r * ldd + c] = sr[(size_t)r * lds_ + c];
    di[(size_t)r * ldd + c] = si[(size_t)r * lds_ + c];
}

// X_block(k0) = Minv(64x64) * X_block(k0), one block per RHS
__global__ __launch_bounds__(NB)
void apply_inv64(const float* __restrict__ Minv,
                 float* __restrict__ Xr, float* __restrict__ Xi, int k0) {
    __shared__ float xr[NB], xi[NB];
    int rhs = blockIdx.x;
    int t = threadIdx.x;
    float* xrb = Xr + (size_t)rhs * NTOT + k0;
    float* xib = Xi + (size_t)rhs * NTOT + k0;
    xr[t] = xrb[t];
    xi[t] = xib[t];
    __syncthreads();
    float ar = 0.0f, ai = 0.0f;
    for (int j = 0; j < NB; ++j) {
        float mr = Minv[t * NB + j], mi = Minv[NB * NB + t * NB + j];
        ar += mr * xr[j] - mi * xi[j];
        ai += mr * xi[j] + mi * xr[j];
    }
    xrb[t] = ar;
    xib[t] = ai;
}

// X[rows row_base..row_base+nrows) -= A[row, k0:k0+64] * X_block(k0), 8 RHS
__global__ __launch_bounds__(256)
void panel_update(const float* __restrict__ Ar, const float* __restrict__ Ai, int k0,
                  float* __restrict__ Xr, float* __restrict__ Xi,
                  int row_base, int nrows) {
    __shared__ float xr[NRHS][NB], xi[NRHS][NB];
    int t = threadIdx.x;
    for (int e = t; e < NRHS * NB; e += 256) {
        int r = e / NB, j = e % NB;
        xr[r][j] = Xr[(size_t)r * NTOT + k0 + j];
        xi[r][j] = Xi[(size_t)r * NTOT + k0 + j];
    }
    __syncthreads();
    int row = row_base + blockIdx.x * 256 + t;
    if (row >= row_base + nrows) return;
    const float* arow_r = Ar + (size_t)row * NTOT + k0;
    const float* arow_i = Ai + (size_t)row * NTOT + k0;
    float accr[NRHS] = {}, acci[NRHS] = {};
    for (int j = 0; j < NB; ++j) {
        float a_r = arow_r[j], a_i = arow_i[j];
#pragma unroll
        for (int r = 0; r < NRHS; ++r) {
            accr[r] += a_r * xr[r][j] - a_i * xi[r][j];
            acci[r] += a_r * xi[r][j] + a_i * xr[r][j];
        }
    }
#pragma unroll
    for (int r = 0; r < NRHS; ++r) {
        Xr[(size_t)r * NTOT + row] -= accr[r];
        Xi[(size_t)r * NTOT + row] -= acci[r];
    }
}

// overlaps[b,o] = |sum conj(Ez_b) * probe_o| -> sigmoid(-(|ov|-E0)*alpha)
__global__ __launch_bounds__(256)
void overlap_out(const float* __restrict__ Xr, const float* __restrict__ Xi,
                 const float* __restrict__ probes, const float* __restrict__ E0s,
                 const float* __restrict__ alpha, float* __restrict__ out) {
    __shared__ float sre[256], sim[256];
    int b = blockIdx.x >> 1, o = blockIdx.x & 1;
    int t = threadIdx.x;
    float ar = 0.0f, ai = 0.0f;
    for (int i = t; i < NTOT; i += 256) {
        float p = probes[o * NTOT + i];
        ar += Xr[(size_t)b * NTOT + i] * p;
        ai -= Xi[(size_t)b * NTOT + i] * p;
    }
    sre[t] = ar; sim[t] = ai;
    __syncthreads();
    for (int s = 128; s > 0; s >>= 1) {
        if (t < s) { sre[t] += sre[t + s]; sim[t] += sim[t + s]; }
        __syncthreads();
    }
    if (t == 0) {
        float mag = sqrtf(sre[0] * sre[0] + sim[0] * sim[0]);
        float z = (mag - E0s[o]) * alpha[0];
        out[b * 2 + o] = 1.0f / (1.0f + expf(z));
    }
}

// ---------------------------------------------------------------------------
extern "C" void kernel_launch(void* const* d_in, const int* in_sizes, int n_in,
                              void* d_out, int out_size, void* d_ws, size_t ws_size,
                              hipStream_t stream) {
    const float* masks  = (const float*)d_in[0];
    const float* rho    = (const float*)d_in[1];
    const float* E0s    = (const float*)d_in[2];
    const float* alpha  = (const float*)d_in[3];
    const float* ics    = (const float*)d_in[4];
    const float* probes = (const float*)d_in[5];
    const float* bg     = (const float*)d_in[6];
    const float* opt    = (const float*)d_in[7];
    float* out = (float*)d_out;

    // Workspace carving (floats)
    const size_t NN = (size_t)NTOT * NTOT;
    float* w    = (float*)d_ws;
    float* Ar   = w;                     w += NN;
    float* Ai   = w;                     w += NN;
    float* Linv = w;                     w += (size_t)NBLK * 2 * NB * NB;
    float* Uinv = w;                     w += (size_t)NBLK * 2 * NB * NB;
    const size_t pmax = (size_t)(NTOT - NB) * NB;
    float* Pr = w; w += pmax;            // column-panel temp (rem x 64)
    float* Pi = w; w += pmax;
    float* Qr = w; w += pmax;            // row-panel temp (64 x rem)
    float* Qi = w; w += pmax;
    float* Xr = w; w += (size_t)NRHS * NTOT;
    float* Xi = w; w += (size_t)NRHS * NTOT;

    // 1) Build A and RHS
    {
        long n2 = (long)(2 * NN);
        zero_f<<<(unsigned)((n2 + 255) / 256), 256, 0, stream>>>(Ar, n2);
        build_A<<<(NTOT + 255) / 256, 256, 0, stream>>>(Ar, Ai, rho, bg, opt);
        init_X<<<(NRHS * NTOT + 255) / 256, 256, 0, stream>>>(Xr, Xi, masks, ics);
    }

    // 2) Blocked LU (no pivoting), WMMA trailing updates
    for (int k = 0; k < NBLK; ++k) {
        int k0 = k * NB, k1 = k0 + NB, rem = NTOT - k1;
        float* Lk = Linv + (size_t)k * 2 * NB * NB;
        float* Uk = Uinv + (size_t)k * 2 * NB * NB;

        lu_diag<<<1, 256, 0, stream>>>(Ar, Ai, k0, Lk, Uk);

        if (rem > 0) {
            // L21 = A21 * inv(U11)   -> P (rem x 64, ld 64)
            cgemm_wmma<<<dim3(NB / 32, rem / 32), 128, 0, stream>>>(
                Ar + (size_t)k1 * NTOT + k0, Ai + (size_t)k1 * NTOT + k0, NTOT,
                Uk, Uk + NB * NB, NB,
                Pr, Pi, NB, NB, 1.0f, 0);
            // U12 = inv(L11) * A12   -> Q (64 x rem, ld rem)
            cgemm_wmma<<<dim3(rem / 32, NB / 32), 128, 0, stream>>>(
                Lk, Lk + NB * NB, NB,
                Ar + (size_t)k0 * NTOT + k1, Ai + (size_t)k0 * NTOT + k1, NTOT,
                Qr, Qi, rem, NB, 1.0f, 0);
            // Commit panels into A
            copy_panel<<<(rem * NB + 255) / 256, 256, 0, stream>>>(
                Pr, Pi, NB,
                Ar + (size_t)k1 * NTOT + k0, Ai + (size_t)k1 * NTOT + k0, NTOT,
                rem, NB);
            copy_panel<<<(NB * rem + 255) / 256, 256, 0, stream>>>(
                Qr, Qi, rem,
                Ar + (size_t)k0 * NTOT + k1, Ai + (size_t)k0 * NTOT + k1, NTOT,
                NB, rem);
            // A22 -= L21 * U12
            cgemm_wmma<<<dim3(rem / 32, rem / 32), 128, 0, stream>>>(
                Ar + (size_t)k1 * NTOT + k0, Ai + (size_t)k1 * NTOT + k0, NTOT,
                Ar + (size_t)k0 * NTOT + k1, Ai + (size_t)k0 * NTOT + k1, NTOT,
                Ar + (size_t)k1 * NTOT + k1, Ai + (size_t)k1 * NTOT + k1, NTOT,
                NB, -1.0f, 1);
        }
    }

    // 3) Forward solve  L y = b  (unit lower; saved block inverses)
    for (int k = 0; k < NBLK; ++k) {
        int k0 = k * NB, k1 = k0 + NB, rem = NTOT - k1;
        apply_inv64<<<NRHS, NB, 0, stream>>>(Linv + (size_t)k * 2 * NB * NB, Xr, Xi, k0);
        if (rem > 0)
            panel_update<<<(rem + 255) / 256, 256, 0, stream>>>(Ar, Ai, k0, Xr, Xi, k1, rem);
    }
    // 4) Backward solve  U x = y
    for (int k = NBLK - 1; k >= 0; --k) {
        int k0 = k * NB;
        apply_inv64<<<NRHS, NB, 0, stream>>>(Uinv + (size_t)k * 2 * NB * NB, Xr, Xi, k0);
        if (k0 > 0)
            panel_update<<<(k0 + 255) / 256, 256, 0, stream>>>(Ar, Ai, k0, Xr, Xi, 0, k0);
    }

    // 5) Overlaps + sigmoid -> d_out [8,2]
    overlap_out<<<NRHS * 2, 256, 0, stream>>>(Xr, Xi, probes, E0s, alpha, out);

    (void)in_sizes; (void)n_in; (void)out_size; (void)ws_size;
}